// QPLayer_31885837206238
// MI455X (gfx1250) — compile-verified
//
#include <hip/hip_runtime.h>
#include <math.h>

// ---------------------------------------------------------------------------
// Batched QP (qpth PDIPM, Mehrotra predictor-corrector) for MI455X / gfx1250.
// One workgroup (256 threads, 8 wave32) per batch element; all per-batch state
// lives in LDS. The dense GEMM H = Q + G^T D G runs on V_WMMA_F32_16X16X4_F32
// with one column-block per wave and 8 row-block accumulators held in VGPRs,
// so the D-scaled B fragment is loaded once per K-step and reused by 8
// independent WMMAs (operand LDS traffic cut >2x, WMMAs co-execute with VALU).
// ---------------------------------------------------------------------------

typedef __attribute__((ext_vector_type(2))) float v2f;
typedef __attribute__((ext_vector_type(8))) float v8f;

#define NZi 128   // nz
#define NIi 128   // nineq
#define NEi 32    // neq
#define NBATCH 128
#define MAXIT 20
#define NTHREADS 256

// LDS strides (padded to dodge 64-bank conflicts)
#define GLD 130
#define HLD 130
#define ALD 130
#define RLD 34
#define SLD 33

// ---------------------------------------------------------------------------
// H = Q + G^T diag(d) G  via WMMA f32 16x16x4, then in-place Cholesky (lower).
// Wave w owns column block J = 16*w and all 8 row blocks (8 live accumulators).
// Fragment layouts per CDNA5 ISA 7.12.2:
//   A 16x4 : lane l -> row m=l&15, K pair = 2*(l>>4)+{0,1}
//   B 4x16 : lane l -> col n=l&15, K pair = 2*(l>>4)+{0,1}
//   C 16x16: vgpr r, lane l -> row r+8*(l>>4), col l&15
// ---------------------------------------------------------------------------
__device__ __forceinline__ void build_H_chol(const float* __restrict__ Gs,
                                             float* __restrict__ Hs,
                                             const float* __restrict__ dvv,
                                             const float* __restrict__ Qg,
                                             int tid) {
  const int lane = tid & 31;
  const int wave = tid >> 5;
  const int m  = lane & 15;
  const int hi = lane >> 4;
  const int J  = wave << 4;          // one 16-wide column block per wave

  v8f acc[8];
  #pragma unroll
  for (int Ib = 0; Ib < 8; ++Ib) {
    const int I = Ib << 4;
    #pragma unroll
    for (int r = 0; r < 8; ++r)
      acc[Ib][r] = Qg[(I + r + 8 * hi) * NZi + (J + m)];
  }

  #pragma unroll 1
  for (int k0 = 0; k0 < NIi; k0 += 4) {
    const int kb = k0 + 2 * hi;
    // B = (D G) fragment for this wave's column block: loaded once, used 8x.
    v2f bf;
    bf.x = dvv[kb]     * Gs[kb * GLD + J + m];
    bf.y = dvv[kb + 1] * Gs[(kb + 1) * GLD + J + m];
    #pragma unroll
    for (int Ib = 0; Ib < 8; ++Ib) {
      const int I = Ib << 4;
      v2f a;                          // A = (G^T) fragment: A[m][k] = G[k][I+m]
      a.x = Gs[kb * GLD + I + m];
      a.y = Gs[(kb + 1) * GLD + I + m];
      acc[Ib] = __builtin_amdgcn_wmma_f32_16x16x4_f32(false, a, false, bf,
                                                      (short)0, acc[Ib],
                                                      false, false);
    }
  }

  #pragma unroll
  for (int Ib = 0; Ib < 8; ++Ib) {
    const int I = Ib << 4;
    #pragma unroll
    for (int r = 0; r < 8; ++r)
      Hs[(I + r + 8 * hi) * HLD + (J + m)] = acc[Ib][r];
  }
  __syncthreads();

  // Right-looking Cholesky, lower triangle in place.
  #pragma unroll 1
  for (int k = 0; k < NZi; ++k) {
    if (tid == 0) Hs[k * HLD + k] = sqrtf(Hs[k * HLD + k]);
    __syncthreads();
    const float dk = Hs[k * HLD + k];
    for (int i = k + 1 + tid; i < NZi; i += NTHREADS) Hs[i * HLD + k] /= dk;
    __syncthreads();
    for (int j = k + 1 + tid; j < NZi; j += NTHREADS) {
      const float ljk = Hs[j * HLD + k];
      #pragma unroll 1
      for (int i = j; i < NZi; ++i)
        Hs[i * HLD + j] -= Hs[i * HLD + k] * ljk;
    }
    __syncthreads();
  }
}

// RHS columns: R[:,0..31] = A^T, R[:,32] = r1 = -rx + G^T (rs - d*rz)
__device__ __forceinline__ void build_rhs_affine(const float* __restrict__ Gs,
                                                 const float* __restrict__ As,
                                                 float* __restrict__ R,
                                                 const float* __restrict__ rxA,
                                                 const float* __restrict__ rsA,
                                                 const float* __restrict__ rzA,
                                                 const float* __restrict__ dvv,
                                                 float* __restrict__ tmp,
                                                 int tid) {
  for (int j = tid; j < NIi; j += NTHREADS) tmp[j] = rsA[j] - dvv[j] * rzA[j];
  __syncthreads();
  for (int i = tid; i < NZi; i += NTHREADS) {
    #pragma unroll
    for (int r = 0; r < NEi; ++r) R[i * RLD + r] = As[r * ALD + i];
    float acc = -rxA[i];
    #pragma unroll 1
    for (int j = 0; j < NIi; ++j) acc += Gs[j * GLD + i] * tmp[j];
    R[i * RLD + NEi] = acc;
  }
  __syncthreads();
}

// Solve L L^T X = R[:, colBase .. colBase+ncols-1] in place; thread-per-column.
__device__ __forceinline__ void tri_solve(const float* __restrict__ Hs,
                                          float* __restrict__ R,
                                          int colBase, int ncols, int tid) {
  if (tid < ncols) {
    const int c = colBase + tid;
    #pragma unroll 1
    for (int i = 0; i < NZi; ++i) {
      float acc = R[i * RLD + c];
      for (int j = 0; j < i; ++j) acc -= Hs[i * HLD + j] * R[j * RLD + c];
      R[i * RLD + c] = acc / Hs[i * HLD + i];
    }
    #pragma unroll 1
    for (int i = NZi - 1; i >= 0; --i) {
      float acc = R[i * RLD + c];
      for (int j = i + 1; j < NZi; ++j) acc -= Hs[j * HLD + i] * R[j * RLD + c];
      R[i * RLD + c] = acc / Hs[i * HLD + i];
    }
  }
  __syncthreads();
}

// S = A * M (M = R cols 0..31), then Cholesky of S (32x32).
__device__ __forceinline__ void schur_chol(const float* __restrict__ As,
                                           const float* __restrict__ R,
                                           float* __restrict__ Ss, int tid) {
  for (int e = tid; e < NEi * NEi; e += NTHREADS) {
    const int r = e >> 5, c = e & 31;
    float acc = 0.f;
    #pragma unroll 1
    for (int i = 0; i < NZi; ++i) acc += As[r * ALD + i] * R[i * RLD + c];
    Ss[r * SLD + c] = acc;
  }
  __syncthreads();
  #pragma unroll 1
  for (int k = 0; k < NEi; ++k) {
    if (tid == 0) Ss[k * SLD + k] = sqrtf(Ss[k * SLD + k]);
    __syncthreads();
    const float dk = Ss[k * SLD + k];
    if (tid > k && tid < NEi) Ss[tid * SLD + k] /= dk;
    __syncthreads();
    if (tid > k && tid < NEi) {
      const int j = tid;
      const float ljk = Ss[j * SLD + k];
      for (int i = j; i < NEi; ++i) Ss[i * SLD + j] -= Ss[i * SLD + k] * ljk;
    }
    __syncthreads();
  }
}

// Given w = H^{-1} r1 in R[:,wcol] and M in R[:,0..31]:
//   S dy = A w - r2   (r2 = -ry  => rhs = A w + ry)
//   dx = w - M dy ;  g = G dx ;  ds = -rz - g ;  dz = -rs + d*(rz + g)
__device__ __forceinline__ void back_solve(const float* __restrict__ As,
                                           const float* __restrict__ Gs,
                                           const float* __restrict__ Ss,
                                           const float* __restrict__ R,
                                           int wcol,
                                           const float* __restrict__ rsA,
                                           const float* __restrict__ rzA,   // nullable
                                           const float* __restrict__ ryA,   // nullable
                                           const float* __restrict__ dvv,
                                           float* __restrict__ gv,
                                           float* __restrict__ dxA,
                                           float* __restrict__ dsA,
                                           float* __restrict__ dzA,
                                           float* __restrict__ dyA,
                                           int tid) {
  if (tid < NEi) {
    float acc = ryA ? ryA[tid] : 0.f;
    #pragma unroll 1
    for (int i = 0; i < NZi; ++i) acc += As[tid * ALD + i] * R[i * RLD + wcol];
    dyA[tid] = acc;
  }
  __syncthreads();
  if (tid == 0) {  // small 32x32 triangular solves
    #pragma unroll 1
    for (int i = 0; i < NEi; ++i) {
      float acc = dyA[i];
      for (int j = 0; j < i; ++j) acc -= Ss[i * SLD + j] * dyA[j];
      dyA[i] = acc / Ss[i * SLD + i];
    }
    #pragma unroll 1
    for (int i = NEi - 1; i >= 0; --i) {
      float acc = dyA[i];
      for (int j = i + 1; j < NEi; ++j) acc -= Ss[j * SLD + i] * dyA[j];
      dyA[i] = acc / Ss[i * SLD + i];
    }
  }
  __syncthreads();
  for (int i = tid; i < NZi; i += NTHREADS) {
    float acc = R[i * RLD + wcol];
    #pragma unroll
    for (int c = 0; c < NEi; ++c) acc -= R[i * RLD + c] * dyA[c];
    dxA[i] = acc;
  }
  __syncthreads();
  for (int j = tid; j < NIi; j += NTHREADS) {
    float acc = 0.f;
    #pragma unroll 1
    for (int i = 0; i < NZi; ++i) acc += Gs[j * GLD + i] * dxA[i];
    gv[j] = acc;
  }
  __syncthreads();
  for (int j = tid; j < NIi; j += NTHREADS) {
    const float rzj = rzA ? rzA[j] : 0.f;
    dsA[j] = -rzj - gv[j];
    dzA[j] = -rsA[j] + dvv[j] * (rzj + gv[j]);
  }
  __syncthreads();
}

__device__ __forceinline__ float get_step_min(const float* __restrict__ v,
                                              const float* __restrict__ dvv, int n) {
  float a = 1e10f;
  #pragma unroll 1
  for (int i = 0; i < n; ++i)
    if (dvv[i] < 0.f) a = fminf(a, -v[i] / dvv[i]);
  return a;
}

constexpr size_t LDS_FLOATS =
    (size_t)NIi * GLD + (size_t)NZi * HLD + (size_t)NEi * ALD +
    (size_t)NZi * RLD + (size_t)NEi * SLD +
    (NZi + NIi + NIi + NEi) +      // x,s,z,y
    (NZi + NIi + NEi) +            // rx,rz,ry
    (NIi + NIi + NIi) +            // d,rs,g
    (NZi + NIi + NIi + NEi) +      // dx,ds,dz,dy
    (NZi + NIi + NIi + NEi) +      // dxa,dsa,dza,dya
    8;                             // scalars

__global__ __launch_bounds__(NTHREADS)
void qp_pdipm_kernel(const float* __restrict__ inputs, const float* __restrict__ Qg,
                     const float* __restrict__ Gg, const float* __restrict__ hg,
                     const float* __restrict__ Ag, const float* __restrict__ bg,
                     float* __restrict__ out) {
  extern __shared__ float lds[];
  float* Gs  = lds;
  float* Hs  = Gs + NIi * GLD;
  float* As  = Hs + NZi * HLD;
  float* R   = As + NEi * ALD;
  float* Ss  = R  + NZi * RLD;
  float* xv  = Ss + NEi * SLD;
  float* sv  = xv + NZi;
  float* zv  = sv + NIi;
  float* yv  = zv + NIi;
  float* rx  = yv + NEi;
  float* rzv = rx + NZi;
  float* ryv = rzv + NIi;
  float* dvv = ryv + NEi;
  float* rsv = dvv + NIi;
  float* gv  = rsv + NIi;
  float* dx_ = gv  + NIi;
  float* ds_ = dx_ + NZi;
  float* dz_ = ds_ + NIi;
  float* dy_ = dz_ + NIi;
  float* dxa = dy_ + NEi;
  float* dsa = dxa + NZi;
  float* dza = dsa + NIi;
  float* dya = dza + NIi;
  float* scal = dya + NEi;

  const int tid = threadIdx.x;
  const int b   = blockIdx.x;
  const float* p = inputs + b * NZi;

  // Stage G, A into LDS (padded strides).
  for (int idx = tid; idx < NIi * NZi; idx += NTHREADS)
    Gs[(idx >> 7) * GLD + (idx & 127)] = Gg[idx];
  for (int idx = tid; idx < NEi * NZi; idx += NTHREADS)
    As[(idx >> 7) * ALD + (idx & 127)] = Ag[idx];
  // Init solve residuals: rx=p, rs=0, rz=-h, ry=-b, d=1.
  for (int i = tid; i < NZi; i += NTHREADS) rx[i] = p[i];
  for (int j = tid; j < NIi; j += NTHREADS) {
    dvv[j] = 1.0f; rsv[j] = 0.0f; rzv[j] = -hg[j];
  }
  for (int k = tid; k < NEi; k += NTHREADS) ryv[k] = -bg[k];
  __syncthreads();

  // ---------------- init KKT solve (D = I) ----------------
  build_H_chol(Gs, Hs, dvv, Qg, tid);
  build_rhs_affine(Gs, As, R, rx, rsv, rzv, dvv, gv, tid);
  tri_solve(Hs, R, 0, NEi + 1, tid);
  schur_chol(As, R, Ss, tid);
  back_solve(As, Gs, Ss, R, NEi, rsv, rzv, ryv, dvv, gv, dx_, ds_, dz_, dy_, tid);

  for (int i = tid; i < NZi; i += NTHREADS) xv[i] = dx_[i];
  for (int j = tid; j < NIi; j += NTHREADS) { sv[j] = ds_[j]; zv[j] = dz_[j]; }
  for (int k = tid; k < NEi; k += NTHREADS) yv[k] = dy_[k];
  __syncthreads();
  if (tid == 0) {
    float ms = 1e30f, mz = 1e30f;
    for (int j = 0; j < NIi; ++j) { ms = fminf(ms, sv[j]); mz = fminf(mz, zv[j]); }
    scal[0] = ms; scal[1] = mz;
  }
  __syncthreads();
  {
    const float ms = scal[0], mz = scal[1];
    for (int j = tid; j < NIi; j += NTHREADS) {
      if (ms < 0.f) sv[j] = sv[j] - ms + 1.0f;
      if (mz < 0.f) zv[j] = zv[j] - mz + 1.0f;
    }
  }
  __syncthreads();

  // ---------------- PDIPM iterations ----------------
  #pragma unroll 1
  for (int it = 0; it < MAXIT; ++it) {
    // Residuals (Q symmetric: x@Q read column-major for coalescing).
    for (int i = tid; i < NZi; i += NTHREADS) {
      float acc = p[i];
      #pragma unroll 1
      for (int j = 0; j < NIi; ++j) acc += zv[j] * Gs[j * GLD + i];
      #pragma unroll 1
      for (int k = 0; k < NEi; ++k) acc += yv[k] * As[k * ALD + i];
      #pragma unroll 1
      for (int j = 0; j < NZi; ++j) acc += Qg[j * NZi + i] * xv[j];
      rx[i] = acc;
    }
    for (int j = tid; j < NIi; j += NTHREADS) {
      float acc = sv[j] - hg[j];
      #pragma unroll 1
      for (int i = 0; i < NZi; ++i) acc += Gs[j * GLD + i] * xv[i];
      rzv[j] = acc;
      rsv[j] = zv[j];
      dvv[j] = zv[j] / sv[j];
    }
    for (int k = tid; k < NEi; k += NTHREADS) {
      float acc = -bg[k];
      #pragma unroll 1
      for (int i = 0; i < NZi; ++i) acc += As[k * ALD + i] * xv[i];
      ryv[k] = acc;
    }
    __syncthreads();
    if (tid == 0) {
      float acc = 0.f;
      for (int j = 0; j < NIi; ++j) acc += sv[j] * zv[j];
      scal[0] = acc;
    }
    __syncthreads();
    const float szsum = scal[0];
    const float mu = szsum * (1.0f / NIi);

    // Factor + affine (predictor) step.
    build_H_chol(Gs, Hs, dvv, Qg, tid);
    build_rhs_affine(Gs, As, R, rx, rsv, rzv, dvv, gv, tid);
    tri_solve(Hs, R, 0, NEi + 1, tid);
    schur_chol(As, R, Ss, tid);
    back_solve(As, Gs, Ss, R, NEi, rsv, rzv, ryv, dvv, gv, dxa, dsa, dza, dya, tid);

    // Centering parameter.
    if (tid == 0) {
      float t = fminf(fminf(get_step_min(zv, dza, NIi),
                            get_step_min(sv, dsa, NIi)), 1.0f);
      float num = 0.f;
      for (int j = 0; j < NIi; ++j)
        num += (sv[j] + t * dsa[j]) * (zv[j] + t * dza[j]);
      float sig = num / szsum;
      sig = sig * sig * sig;
      scal[1] = mu * sig;
    }
    __syncthreads();
    const float musig = scal[1];
    for (int j = tid; j < NIi; j += NTHREADS)
      rsv[j] = (dsa[j] * dza[j] - musig) / sv[j];     // rs_c (rsv no longer needed as z)
    __syncthreads();

    // Corrector: rx=rz=ry=0 => r1_c = G^T rs_c, reuse L and M.
    for (int i = tid; i < NZi; i += NTHREADS) {
      float acc = 0.f;
      #pragma unroll 1
      for (int j = 0; j < NIi; ++j) acc += Gs[j * GLD + i] * rsv[j];
      R[i * RLD + NEi + 1] = acc;
    }
    __syncthreads();
    tri_solve(Hs, R, NEi + 1, 1, tid);
    back_solve(As, Gs, Ss, R, NEi + 1, rsv, nullptr, nullptr, dvv, gv,
               dx_, ds_, dz_, dy_, tid);

    // Combine, step length, update.
    for (int i = tid; i < NZi; i += NTHREADS) dx_[i] += dxa[i];
    for (int j = tid; j < NIi; j += NTHREADS) { ds_[j] += dsa[j]; dz_[j] += dza[j]; }
    for (int k = tid; k < NEi; k += NTHREADS) dy_[k] += dya[k];
    __syncthreads();
    if (tid == 0) {
      float al = fminf(0.999f * fminf(get_step_min(sv, ds_, NIi),
                                      get_step_min(zv, dz_, NIi)), 1.0f);
      scal[2] = al;
    }
    __syncthreads();
    const float al = scal[2];
    for (int i = tid; i < NZi; i += NTHREADS) xv[i] += al * dx_[i];
    for (int j = tid; j < NIi; j += NTHREADS) { sv[j] += al * ds_[j]; zv[j] += al * dz_[j]; }
    for (int k = tid; k < NEi; k += NTHREADS) yv[k] += al * dy_[k];
    __syncthreads();
  }

  for (int i = tid; i < NZi; i += NTHREADS) out[b * NZi + i] = xv[i];
}

extern "C" void kernel_launch(void* const* d_in, const int* in_sizes, int n_in,
                              void* d_out, int out_size, void* d_ws, size_t ws_size,
                              hipStream_t stream) {
  (void)in_sizes; (void)n_in; (void)d_ws; (void)ws_size; (void)out_size;
  const float* inputs = (const float*)d_in[0];
  const float* Q      = (const float*)d_in[1];
  const float* G      = (const float*)d_in[2];
  const float* h      = (const float*)d_in[3];
  const float* A      = (const float*)d_in[4];
  const float* b      = (const float*)d_in[5];
  float* out = (float*)d_out;
  qp_pdipm_kernel<<<dim3(NBATCH), dim3(NTHREADS),
                    LDS_FLOATS * sizeof(float), stream>>>(inputs, Q, G, h, A, b, out);
}